// ResBlockMoE_65163243815191
// MI455X (gfx1250) — compile-verified
//
#include <hip/hip_runtime.h>
#include <hip/hip_bf16.h>

typedef float v2f __attribute__((ext_vector_type(2)));
typedef float v8f __attribute__((ext_vector_type(8)));
typedef unsigned int u32x4_t __attribute__((ext_vector_type(4)));
typedef int i32x4_t __attribute__((ext_vector_type(4)));
typedef int i32x8_t __attribute__((ext_vector_type(8)));

#define BIGF 3.0e38f

// ---------------------------------------------------------------------------
// WMMA helper: D = A(16x4,f32) x B(4x16,f32) + C(16x16,f32), wave32.
// A frag (2 VGPR): lanes 0-15 hold (K0,K1) for M=lane; lanes 16-31 hold (K2,K3).
// B frag (2 VGPR): lanes 0-15 hold (K0row,K1row) at N=lane; lanes 16-31 (K2,K3).
// C/D  (8 VGPR): vgpr r -> M = r + 8*(lane>=16), N = lane&15.
// ---------------------------------------------------------------------------
__device__ __forceinline__ v8f wmma4(v2f a, v2f b, v8f c) {
  return __builtin_amdgcn_wmma_f32_16x16x4_f32(false, a, false, b, (short)0, c,
                                               false, false);
}

// ---------------------------------------------------------------------------
// Tensor Data Mover: 1-D DMA of `nelem` f32 from global to LDS (TENSORcnt).
// D# built per CDNA5 ISA: group0 = count/lds/global/type,
// group1 = data_size + tensor_dim0/tile_dim0; groups 2/3 unused (zero).
// ---------------------------------------------------------------------------
__device__ __forceinline__ void tdm_load_1d(unsigned lds_byte_off,
                                            const float* gptr, unsigned nelem) {
  unsigned long long ga = (unsigned long long)(uintptr_t)gptr;
  u32x4_t g0;
  g0.x = 1u;                                              // count=1 (user D#)
  g0.y = lds_byte_off;                                    // lds_addr
  g0.z = (unsigned)ga;                                    // global_addr[31:0]
  g0.w = ((unsigned)(ga >> 32) & 0x01FFFFFFu) | (2u << 30); // addr[56:32]|type=2
  i32x8_t g1;
  g1[0] = (2 << 16);                                      // data_size = 4 bytes
  g1[1] = (int)((nelem & 0xFFFFu) << 16);                 // tensor_dim0[15:0]
  g1[2] = (int)((nelem >> 16) & 0xFFFFu) | (1 << 16);     // dim0[31:16], dim1=1
  g1[3] = (int)((nelem & 0xFFFFu) << 16);                 // tile_dim0 = nelem
  g1[4] = 0;                                              // tile_dim1/2 unused
  g1[5] = (int)nelem;                                     // dim0_stride[31:0]
  g1[6] = 0;
  g1[7] = 0;
  i32x4_t z4 = (i32x4_t)0;
#if __clang_major__ >= 23
  i32x8_t z8 = (i32x8_t)0;
  __builtin_amdgcn_tensor_load_to_lds(g0, g1, z4, z4, z8, 0);
#else
  __builtin_amdgcn_tensor_load_to_lds(g0, g1, z4, z4, 0);
#endif
}

__device__ __forceinline__ unsigned lds_off(const void* p) {
  return (unsigned)(uintptr_t)p;   // shared aperture: low 32 bits = LDS offset
}

__device__ __forceinline__ float siluf(float v) { return v / (1.f + expf(-v)); }

// Block-wide reductions (blockDim.x == 256)
__device__ __forceinline__ float bred_sum(float v, float* s) {
  int t = threadIdx.x; s[t] = v; __syncthreads();
  #pragma unroll
  for (int k = 128; k > 0; k >>= 1) { if (t < k) s[t] += s[t + k]; __syncthreads(); }
  float r = s[0]; __syncthreads(); return r;
}
__device__ __forceinline__ float bred_min(float v, float* s) {
  int t = threadIdx.x; s[t] = v; __syncthreads();
  #pragma unroll
  for (int k = 128; k > 0; k >>= 1) { if (t < k) s[t] = fminf(s[t], s[t + k]); __syncthreads(); }
  float r = s[0]; __syncthreads(); return r;
}
__device__ __forceinline__ float bred_max(float v, float* s) {
  int t = threadIdx.x; s[t] = v; __syncthreads();
  #pragma unroll
  for (int k = 128; k > 0; k >>= 1) { if (t < k) s[t] = fmaxf(s[t], s[t + k]); __syncthreads(); }
  float r = s[0]; __syncthreads(); return r;
}

// ---------------------------------------------------------------------------
// K0: partial global min/max of x
// ---------------------------------------------------------------------------
__global__ void k_minmax_part(const float* __restrict__ x, int n,
                              float* __restrict__ pmin, float* __restrict__ pmax) {
  __shared__ float sb[256];
  float mn = BIGF, mx = -BIGF;
  for (int i = blockIdx.x * 256 + threadIdx.x; i < n; i += gridDim.x * 256) {
    float v = x[i]; mn = fminf(mn, v); mx = fmaxf(mx, v);
  }
  float MN = bred_min(mn, sb);
  float MX = bred_max(mx, sb);
  if (threadIdx.x == 0) { pmin[blockIdx.x] = MN; pmax[blockIdx.x] = MX; }
}

// K0b: final min/max -> xn affine consts: xn = a*x + b
__global__ void k_minmax_fin(const float* __restrict__ pmin,
                             const float* __restrict__ pmax, int nb,
                             float* __restrict__ nc) {
  __shared__ float sb[256];
  float mn = BIGF, mx = -BIGF;
  for (int i = threadIdx.x; i < nb; i += 256) {
    mn = fminf(mn, pmin[i]); mx = fmaxf(mx, pmax[i]);
  }
  float MN = bred_min(mn, sb);
  float MX = bred_max(mx, sb);
  if (threadIdx.x == 0) {
    float d = MX - MN;
    nc[0] = 2.f / d;
    nc[1] = -2.f * MN / d - 1.f;
  }
}

// ---------------------------------------------------------------------------
// K1: pack weights into WMMA A-fragment layout, contiguous per 64-row M-group
// so each GEMM block's K-chunk is a flat TDM copy.
// Source: base (E,M,C,T), poly (E,M,C*4,T) with poly channel = d*C + c.
// K-slices per (c,t): k0=base, k1=poly(d1), k2=poly(d2), k3=poly(d3).
// Out: [(((e*T+t)*MG + mg)*C + c)*4 + mtl][lane] = float2, mg=mt>>2, mtl=mt&3.
// ---------------------------------------------------------------------------
__global__ void k_pack(const float* __restrict__ base, const float* __restrict__ poly,
                       float* __restrict__ out, int E, int M, int C, int T, int total) {
  int idx = blockIdx.x * 256 + threadIdx.x;
  if (idx >= total) return;
  int lane = idx & 31; int r = idx >> 5;
  int MT = M >> 4, MG = M >> 6;
  int mt = r % MT; r /= MT;
  int c  = r % C;  r /= C;
  int t  = r % T;  int e = r / T;
  int m  = mt * 16 + (lane & 15);
  int mg = mt >> 2, mtl = mt & 3;
  int k0 = (lane < 16) ? 0 : 2;
  float v0 = (k0 == 0) ? base[((e * M + m) * C + c) * T + t]
                       : poly[((e * M + m) * (C * 4) + k0 * C + c) * T + t];
  float v1 = poly[((e * M + m) * (C * 4) + (k0 + 1) * C + c) * T + t];
  long long o = ((((long long)(e * T + t) * MG + mg) * C + c) * 4 + mtl) * 64 + lane * 2;
  out[o] = v0; out[o + 1] = v1;
}

// K1b: P0 (=1) fold -> bias[m] = sum_c poly[m][0*C + c]  (1x1 layers, T=1)
__global__ void k_bias(const float* __restrict__ poly, int M, int C,
                       float* __restrict__ bias) {
  int m = blockIdx.x * 256 + threadIdx.x;
  if (m >= M) return;
  float s = 0.f;
  for (int c = 0; c < C; ++c) s += poly[m * (C * 4) + c];
  bias[m] = s;
}

// K1c: MoE P0 partial bias per tap: se[(e*3+t)*64+w] = sum_c poly[e,w,c,t]
__global__ void k_se(const float* __restrict__ poly, float* __restrict__ se) {
  int i = blockIdx.x * 256 + threadIdx.x;
  if (i >= 8 * 3 * 64) return;
  int w = i & 63; int t = (i >> 6) % 3; int e = i / 192;
  float s = 0.f;
  for (int c = 0; c < 64; ++c) s += poly[((e * 64 + w) * 256 + c) * 3 + t];
  se[(e * 3 + t) * 64 + w] = s;
}

// ---------------------------------------------------------------------------
// K2/K9: fused KALN 1x1 GEMM. z[b,m,l] = bias[m] +
//   sum_c ( W0*x + W1*P1(xn) + W2*P2(xn) + W3*P3(xn) ), xn = a*x+b.
// One wave computes a 64(M) x 16(N) tile, K swept 4-at-a-time (one channel).
// A fragments staged via TDM (double-buffered 32-channel chunks in LDS, each
// wave DMAs a 4KB sub-slice, TENSORcnt-tracked); basis generated in registers.
// grid = (Ntiles/8, Mtot/64), block 256.
// ---------------------------------------------------------------------------
__global__ void k_gemm1x1(const float* __restrict__ in,     // (B, Cin, 2048)
                          const float* __restrict__ apack,  // packed A frags
                          const float* __restrict__ nc,     // a, b
                          const float* __restrict__ bias,   // [Mtot]
                          float* __restrict__ z,            // (B, Mtot, 2048)
                          int Cin, int Mtot) {
  __shared__ float ldsA[2][8192];                           // 2 x 32KB
  const int lane = threadIdx.x & 31;
  const int wv   = threadIdx.x >> 5;
  const int nt   = blockIdx.x * 8 + wv;
  const int n0   = nt * 16;
  const int b    = n0 >> 11;
  const int l    = (n0 & 2047) + (lane & 15);
  const int mg   = blockIdx.y;
  const float a  = nc[0], bb = nc[1];
  const bool lo  = lane < 16;
  const float* xin   = in + ((long long)b * Cin) * 2048 + l;
  const float* abase = apack + ((long long)mg * Cin) * 256;
  const int nch = Cin >> 5;                                 // 32-channel chunks

  // prime chunk 0: this wave DMAs channels [wv*4, wv*4+4) of the chunk
  tdm_load_1d(lds_off(&ldsA[0][0]) + wv * 4096, abase + wv * 1024, 1024);

  v8f acc[4] = {};
  for (int k = 0; k < nch; ++k) {
    if (k + 1 < nch) {
      tdm_load_1d(lds_off(&ldsA[(k + 1) & 1][0]) + wv * 4096,
                  abase + (long long)(k + 1) * 8192 + wv * 1024, 1024);
      __builtin_amdgcn_s_wait_tensorcnt(1);
    } else {
      __builtin_amdgcn_s_wait_tensorcnt(0);
    }
    __syncthreads();                                        // chunk k visible
    const float* Ab = &ldsA[k & 1][0];
    for (int cc = 0; cc < 32; ++cc) {
      const int c = k * 32 + cc;
      float xv = xin[(long long)c * 2048];
      __builtin_prefetch(xin + (long long)(c + 8) * 2048, 0, 3);
      float xn = fmaf(a, xv, bb);
      float p2 = fmaf(1.5f * xn, xn, -0.5f);
      float p3 = (5.f * xn * p2 - 2.f * xn) * (1.f / 3.f);
      v2f bf; bf.x = lo ? xv : p2; bf.y = lo ? xn : p3;
      const float* ap = Ab + cc * 256 + lane * 2;
      v2f a0 = *(const v2f*)(ap);
      v2f a1 = *(const v2f*)(ap + 64);
      v2f a2 = *(const v2f*)(ap + 128);
      v2f a3 = *(const v2f*)(ap + 192);
      acc[0] = wmma4(a0, bf, acc[0]);
      acc[1] = wmma4(a1, bf, acc[1]);
      acc[2] = wmma4(a2, bf, acc[2]);
      acc[3] = wmma4(a3, bf, acc[3]);
    }
    __syncthreads();                                        // buffer reuse safe
  }
  const int mrow = lo ? 0 : 8;
  #pragma unroll
  for (int mt = 0; mt < 4; ++mt) {
    int m0 = mg * 64 + mt * 16 + mrow;
    float* zp = z + ((long long)b * Mtot + m0) * 2048 + l;
    #pragma unroll
    for (int r = 0; r < 8; ++r) zp[(long long)r * 2048] = acc[mt][r] + bias[m0 + r];
  }
}

// ---------------------------------------------------------------------------
// K3: row instance-norm stats + SiLU for layer 1; also gate_in (row mean of h)
// and per-row min/max of h. One block per (b,w) row.
// ---------------------------------------------------------------------------
__global__ void k_rownorm1(const float* __restrict__ z, float* __restrict__ h,
                           float* __restrict__ gate_in, float* __restrict__ rmn,
                           float* __restrict__ rmx) {
  __shared__ float sb[256];
  const int row = blockIdx.x;
  const float* zp = z + (long long)row * 2048;
  float s = 0.f, q = 0.f;
  for (int i = threadIdx.x; i < 2048; i += 256) { float v = zp[i]; s += v; q = fmaf(v, v, q); }
  float S = bred_sum(s, sb), Q = bred_sum(q, sb);
  float mean = S * (1.f / 2048.f);
  float var  = Q * (1.f / 2048.f) - mean * mean;
  float inv  = rsqrtf(var + 1e-5f);
  float hs = 0.f, mn = BIGF, mx = -BIGF;
  float* hp = h + (long long)row * 2048;
  for (int i = threadIdx.x; i < 2048; i += 256) {
    float v = (zp[i] - mean) * inv;
    float hv = siluf(v);
    hp[i] = hv; hs += hv; mn = fminf(mn, hv); mx = fmaxf(mx, hv);
  }
  float HS = bred_sum(hs, sb);
  float MN = bred_min(mn, sb);
  float MX = bred_max(mx, sb);
  if (threadIdx.x == 0) { gate_in[row] = HS * (1.f / 2048.f); rmn[row] = MN; rmx[row] = MX; }
}

// ---------------------------------------------------------------------------
// K4: gating (single block). softmax(gate_in @ w_gate), top-2, gates, CV^2
// loss, per-expert min/max -> per-pair (expert, gate, a, b).
// ---------------------------------------------------------------------------
__global__ void k_gating(const float* __restrict__ gate_in, const float* __restrict__ wg,
                         const float* __restrict__ rmn, const float* __restrict__ rmx,
                         float* __restrict__ pairbuf, float* __restrict__ loss_out) {
  __shared__ float gi[16 * 64];
  __shared__ float bmin[16], bmax[16];
  __shared__ float probs[16 * 8];
  __shared__ float gates[16 * 8];
  __shared__ int   tsel[16 * 2];
  __shared__ float emin[8], emax[8];
  const int tid = threadIdx.x;
  for (int i = tid; i < 1024; i += 256) gi[i] = gate_in[i];
  __syncthreads();
  if (tid < 16) {
    float mn = BIGF, mx = -BIGF;
    for (int w = 0; w < 64; ++w) { mn = fminf(mn, rmn[tid * 64 + w]); mx = fmaxf(mx, rmx[tid * 64 + w]); }
    bmin[tid] = mn; bmax[tid] = mx;
  }
  if (tid < 128) {
    int b = tid >> 3, e = tid & 7;
    float acc = 0.f;
    for (int w = 0; w < 64; ++w) acc += gi[b * 64 + w] * wg[w * 8 + e];
    probs[tid] = acc;
  }
  __syncthreads();
  if (tid < 16) {
    float mx = -BIGF;
    for (int e = 0; e < 8; ++e) mx = fmaxf(mx, probs[tid * 8 + e]);
    float s = 0.f;
    for (int e = 0; e < 8; ++e) { float ev = expf(probs[tid * 8 + e] - mx); probs[tid * 8 + e] = ev; s += ev; }
    float is = 1.f / s;
    float p[8];
    for (int e = 0; e < 8; ++e) { p[e] = probs[tid * 8 + e] * is; }
    int i0 = 0; float v0 = -1.f;
    for (int e = 0; e < 8; ++e) if (p[e] > v0) { v0 = p[e]; i0 = e; }
    p[i0] = -2.f;
    int i1 = 0; float v1 = -1.f;
    for (int e = 0; e < 8; ++e) if (p[e] > v1) { v1 = p[e]; i1 = e; }
    float denom = 1.f / (v0 + v1 + 1e-6f);
    for (int e = 0; e < 8; ++e) gates[tid * 8 + e] = 0.f;
    gates[tid * 8 + i0] = v0 * denom;
    gates[tid * 8 + i1] = v1 * denom;
    tsel[tid * 2] = i0; tsel[tid * 2 + 1] = i1;
  }
  __syncthreads();
  if (tid < 8) {
    float mn = BIGF, mx = -BIGF;
    for (int b = 0; b < 16; ++b)
      if (gates[b * 8 + tid] > 0.f) { mn = fminf(mn, bmin[b]); mx = fmaxf(mx, bmax[b]); }
    emin[tid] = mn; emax[tid] = mx;   // unselected expert: BIG/-BIG, matches ref
  }
  __syncthreads();
  if (tid < 32) {
    int b = tid >> 1, k = tid & 1;
    int e = tsel[b * 2 + k];
    float d = emax[e] - emin[e];
    pairbuf[tid * 4 + 0] = (float)e;
    pairbuf[tid * 4 + 1] = gates[b * 8 + e];
    pairbuf[tid * 4 + 2] = 2.f / d;
    pairbuf[tid * 4 + 3] = -2.f * emin[e] / d - 1.f;
  }
  if (tid == 0) {
    float imp[8], ld[8], im = 0.f, lm = 0.f;
    for (int e = 0; e < 8; ++e) {
      float s = 0.f, c = 0.f;
      for (int b = 0; b < 16; ++b) { float g = gates[b * 8 + e]; s += g; if (g > 0.f) c += 1.f; }
      imp[e] = s; ld[e] = c; im += s; lm += c;
    }
    im *= 0.125f; lm *= 0.125f;
    float vi = 0.f, vl = 0.f;
    for (int e = 0; e < 8; ++e) {
      vi += (imp[e] - im) * (imp[e] - im);
      vl += (ld[e] - lm) * (ld[e] - lm);
    }
    vi *= (1.f / 7.f); vl *= (1.f / 7.f);
    loss_out[0] = 0.01f * (vi / (im * im + 1e-10f) + vl / (lm * lm + 1e-10f));
  }
}

// ---------------------------------------------------------------------------
// K6: MoE KALN K=3 conv as WMMA GEMM (only the 32 selected (b,expert) pairs).
// K = 3 taps x 64 ch x 4 streams; 6 TDM-staged chunks (tap x 32-channel half);
// padded positions supply all-zero streams (basis padded with zeros in ref).
// P0 fold -> tap-partial biases with edge correction at l=0 / l=2047.
// grid = (16, 32 pairs), block 256 (8 waves x one 64x16 tile).
// ---------------------------------------------------------------------------
__global__ void k_moe_gemm(const float* __restrict__ h, const float* __restrict__ apackE,
                           const float* __restrict__ pair, const float* __restrict__ se,
                           float* __restrict__ zE) {
  __shared__ float ldsA[2][8192];
  const int lane = threadIdx.x & 31;
  const int wv   = threadIdx.x >> 5;
  const int p    = blockIdx.y;
  const int e    = (int)pair[p * 4 + 0];
  const float a  = pair[p * 4 + 2];
  const float bb = pair[p * 4 + 3];
  const int b    = p >> 1;
  const int l    = (blockIdx.x * 8 + wv) * 16 + (lane & 15);
  const bool lo  = lane < 16;
  const float* hb = h + (long long)b * 64 * 2048;
  const float* aE = apackE + (long long)e * 3 * 64 * 256;   // this expert

  // chunk k (0..5): tap t = k>>1, channels c0 = (k&1)*32
  tdm_load_1d(lds_off(&ldsA[0][0]) + wv * 4096, aE + wv * 1024, 1024);

  v8f acc[4] = {};
  for (int k = 0; k < 6; ++k) {
    if (k + 1 < 6) {
      tdm_load_1d(lds_off(&ldsA[(k + 1) & 1][0]) + wv * 4096,
                  aE + (long long)(k + 1) * 8192 + wv * 1024, 1024);
      __builtin_amdgcn_s_wait_tensorcnt(1);
    } else {
      __builtin_amdgcn_s_wait_tensorcnt(0);
    }
    __syncthreads();
    const int t  = k >> 1;
    const int c0 = (k & 1) * 32;
    const int lt = l + t - 1;
    const bool valid = (lt >= 0) && (lt < 2048);
    const float* Ab = &ldsA[k & 1][0];
    for (int cc = 0; cc < 32; ++cc) {
      const int c = c0 + cc;
      float hv = valid ? hb[c * 2048 + lt] : 0.f;
      float hn = fmaf(a, hv, bb);
      float p2 = fmaf(1.5f * hn, hn, -0.5f);
      float p3 = (5.f * hn * p2 - 2.f * hn) * (1.f / 3.f);
      v2f bf;
      bf.x = valid ? (lo ? hv : p2) : 0.f;
      bf.y = valid ? (lo ? hn : p3) : 0.f;
      const float* ap = Ab + cc * 256 + lane * 2;
      v2f a0 = *(const v2f*)(ap);
      v2f a1 = *(const v2f*)(ap + 64);
      v2f a2 = *(const v2f*)(ap + 128);
      v2f a3 = *(const v2f*)(ap + 192);
      acc[0] = wmma4(a0, bf, acc[0]);
      acc[1] = wmma4(a1, bf, acc[1]);
      acc[2] = wmma4(a2, bf, acc[2]);
      acc[3] = wmma4(a3, bf, acc[3]);
    }
    __syncthreads();
  }
  const int mrow = lo ? 0 : 8;
  #pragma unroll
  for (int mt = 0; mt < 4; ++mt) {
    #pragma unroll
    for (int r = 0; r < 8; ++r) {
      int w = mt * 16 + mrow + r;
      float s0 = se[(e * 3 + 0) * 64 + w];
      float s1 = se[(e * 3 + 1) * 64 + w];
      float s2 = se[(e * 3 + 2) * 64 + w];
      float bias = s0 + s1 + s2 - (l == 0 ? s0 : 0.f) - (l == 2047 ? s2 : 0.f);
      zE[((long long)p * 64 + w) * 2048 + l] = acc[mt][r] + bias;
    }
  }
}

// ---------------------------------------------------------------------------
// K7: per (b,w): instance-norm + SiLU both selected experts, gate-weighted
// sum -> y; row min/max of y for layer-3 range.
// ---------------------------------------------------------------------------
__global__ void k_combine(const float* __restrict__ zE, const float* __restrict__ pair,
                          float* __restrict__ y, float* __restrict__ rmn,
                          float* __restrict__ rmx) {
  __shared__ float sb[256];
  const int row = blockIdx.x;          // b*64 + w
  const int b = row >> 6, w = row & 63;
  const float* r0 = zE + ((long long)(2 * b) * 64 + w) * 2048;
  const float* r1 = zE + ((long long)(2 * b + 1) * 64 + w) * 2048;
  const float g0 = pair[(2 * b) * 4 + 1];
  const float g1 = pair[(2 * b + 1) * 4 + 1];
  float s0 = 0.f, q0 = 0.f, s1 = 0.f, q1 = 0.f;
  for (int i = threadIdx.x; i < 2048; i += 256) {
    float a = r0[i]; s0 += a; q0 = fmaf(a, a, q0);
    float c = r1[i]; s1 += c; q1 = fmaf(c, c, q1);
  }
  float S0 = bred_sum(s0, sb), Q0 = bred_sum(q0, sb);
  float S1 = bred_sum(s1, sb), Q1 = bred_sum(q1, sb);
  float m0 = S0 * (1.f / 2048.f), i0 = rsqrtf(Q0 * (1.f / 2048.f) - m0 * m0 + 1e-5f);
  float m1 = S1 * (1.f / 2048.f), i1 = rsqrtf(Q1 * (1.f / 2048.f) - m1 * m1 + 1e-5f);
  float mn = BIGF, mx = -BIGF;
  float* yp = y + (long long)row * 2048;
  for (int i = threadIdx.x; i < 2048; i += 256) {
    float v0 = siluf((r0[i] - m0) * i0);
    float v1 = siluf((r1[i] - m1) * i1);
    float yv = g0 * v0 + g1 * v1;
    yp[i] = yv; mn = fminf(mn, yv); mx = fmaxf(mx, yv);
  }
  float MN = bred_min(mn, sb);
  float MX = bred_max(mx, sb);
  if (threadIdx.x == 0) { rmn[row] = MN; rmx[row] = MX; }
}

// ---------------------------------------------------------------------------
// K10: final row norm + SiLU + residual -> d_out
// ---------------------------------------------------------------------------
__global__ void k_rownorm3(const float* __restrict__ z, const float* __restrict__ x,
                           float* __restrict__ out) {
  __shared__ float sb[256];
  const int row = blockIdx.x;          // b*256 + o
  const float* zp = z + (long long)row * 2048;
  float s = 0.f, q = 0.f;
  for (int i = threadIdx.x; i < 2048; i += 256) { float v = zp[i]; s += v; q = fmaf(v, v, q); }
  float S = bred_sum(s, sb), Q = bred_sum(q, sb);
  float mean = S * (1.f / 2048.f);
  float inv  = rsqrtf(Q * (1.f / 2048.f) - mean * mean + 1e-5f);
  const float* xp = x + (long long)row * 2048;
  float* op = out + (long long)row * 2048;
  for (int i = threadIdx.x; i < 2048; i += 256)
    op[i] = siluf((zp[i] - mean) * inv) + xp[i];
}

// ---------------------------------------------------------------------------
// Workspace layout (floats)
// ---------------------------------------------------------------------------
static const long long OFF_H      = 0;                         // 16*64*2048
static const long long OFF_Z1Y    = OFF_H      + 2097152;      // z1, reused as y
static const long long OFF_ZE     = OFF_Z1Y    + 2097152;      // 32*64*2048
static const long long OFF_Z3     = OFF_ZE     + 4194304;      // 16*256*2048
static const long long OFF_AP1    = OFF_Z3     + 8388608;      // 256*4*32*2
static const long long OFF_AP3    = OFF_AP1    + 65536;        // 64*16*32*2
static const long long OFF_APE    = OFF_AP3    + 65536;        // 8*3*64*4*32*2
static const long long OFF_BIAS1  = OFF_APE    + 393216;       // 64
static const long long OFF_BIAS3  = OFF_BIAS1  + 64;           // 256
static const long long OFF_SE     = OFF_BIAS3  + 256;          // 1536
static const long long OFF_PMIN   = OFF_SE     + 1536;         // 512
static const long long OFF_PMAX   = OFF_PMIN   + 512;          // 512
static const long long OFF_NC1    = OFF_PMAX   + 512;          // 2
static const long long OFF_RMIN   = OFF_NC1    + 2;            // 1024
static const long long OFF_RMAX   = OFF_RMIN   + 1024;         // 1024
static const long long OFF_GATEIN = OFF_RMAX   + 1024;         // 1024
static const long long OFF_PAIR   = OFF_GATEIN + 1024;         // 32*4
static const long long OFF_RMINY  = OFF_PAIR   + 128;          // 1024
static const long long OFF_RMAXY  = OFF_RMINY  + 1024;         // 1024
static const long long OFF_NC3    = OFF_RMAXY  + 1024;         // 2

extern "C" void kernel_launch(void* const* d_in, const int* in_sizes, int n_in,
                              void* d_out, int out_size, void* d_ws, size_t ws_size,
                              hipStream_t stream) {
  (void)in_sizes; (void)n_in; (void)ws_size;
  const float* x   = (const float*)d_in[0];
  const float* c1b = (const float*)d_in[1];
  const float* c1p = (const float*)d_in[2];
  const float* mbw = (const float*)d_in[3];
  const float* mpw = (const float*)d_in[4];
  const float* wg  = (const float*)d_in[5];
  const float* c3b = (const float*)d_in[6];
  const float* c3p = (const float*)d_in[7];
  float* ws  = (float*)d_ws;
  float* out = (float*)d_out;
  float* loss_out = out + (out_size - 1);

  // --- global min/max of x -> layer-1 normalization consts
  k_minmax_part<<<512, 256, 0, stream>>>(x, 16 * 256 * 2048, ws + OFF_PMIN, ws + OFF_PMAX);
  k_minmax_fin<<<1, 256, 0, stream>>>(ws + OFF_PMIN, ws + OFF_PMAX, 512, ws + OFF_NC1);

  // --- weight packing (A fragments) + P0-fold biases
  k_pack<<<128, 256, 0, stream>>>(c1b, c1p, ws + OFF_AP1, 1, 64, 256, 1, 32768);
  k_pack<<<128, 256, 0, stream>>>(c3b, c3p, ws + OFF_AP3, 1, 256, 64, 1, 32768);
  k_pack<<<768, 256, 0, stream>>>(mbw, mpw, ws + OFF_APE, 8, 64, 64, 3, 196608);
  k_bias<<<1, 256, 0, stream>>>(c1p, 64, 256, ws + OFF_BIAS1);
  k_bias<<<1, 256, 0, stream>>>(c3p, 256, 64, ws + OFF_BIAS3);
  k_se<<<6, 256, 0, stream>>>(mpw, ws + OFF_SE);

  // --- layer 1: fused KALN 1x1 GEMM (WMMA f32 + TDM A-staging), norm + SiLU
  k_gemm1x1<<<dim3(256, 1), 256, 0, stream>>>(x, ws + OFF_AP1, ws + OFF_NC1,
                                              ws + OFF_BIAS1, ws + OFF_Z1Y, 256, 64);
  k_rownorm1<<<1024, 256, 0, stream>>>(ws + OFF_Z1Y, ws + OFF_H, ws + OFF_GATEIN,
                                       ws + OFF_RMIN, ws + OFF_RMAX);

  // --- gating: softmax, top-2, per-expert ranges, CV^2 loss
  k_gating<<<1, 256, 0, stream>>>(ws + OFF_GATEIN, wg, ws + OFF_RMIN, ws + OFF_RMAX,
                                  ws + OFF_PAIR, loss_out);

  // --- MoE: 32 selected (b,expert) KALN K=3 convs (WMMA + TDM), norm+gate sum
  k_moe_gemm<<<dim3(16, 32), 256, 0, stream>>>(ws + OFF_H, ws + OFF_APE,
                                               ws + OFF_PAIR, ws + OFF_SE, ws + OFF_ZE);
  k_combine<<<1024, 256, 0, stream>>>(ws + OFF_ZE, ws + OFF_PAIR, ws + OFF_Z1Y,
                                      ws + OFF_RMINY, ws + OFF_RMAXY);
  k_minmax_fin<<<1, 256, 0, stream>>>(ws + OFF_RMINY, ws + OFF_RMAXY, 1024, ws + OFF_NC3);

  // --- layer 3: fused KALN 1x1 GEMM (WMMA + TDM), then norm + SiLU + residual
  k_gemm1x1<<<dim3(256, 4), 256, 0, stream>>>(ws + OFF_Z1Y, ws + OFF_AP3, ws + OFF_NC3,
                                              ws + OFF_BIAS3, ws + OFF_Z3, 64, 256);
  k_rownorm3<<<4096, 256, 0, stream>>>(ws + OFF_Z3, x, out);
}